// DynamicGraphConstruction_42279658062318
// MI455X (gfx1250) — compile-verified
//
#include <hip/hip_runtime.h>

typedef __attribute__((ext_vector_type(2))) float v2f;
typedef __attribute__((ext_vector_type(8))) float v8f;

#define BN_EPS 1e-5f

// ---------------------------------------------------------------------------
// K1: per-edge dot product of 12-dim embeddings (random gather, L2-resident).
// float4 x3 loads per row (rows are 48B = 16B aligned). Prefetch the
// sequentially streamed index arrays (global_prefetch_b8, CDNA5 path).
// ---------------------------------------------------------------------------
__global__ void dgc_edge_dot(const float* __restrict__ src,
                             const float* __restrict__ dst,
                             const int* __restrict__ gs,
                             const int* __restrict__ gd,
                             float* __restrict__ lik, int n) {
  int stride = gridDim.x * blockDim.x;
  for (int e = blockIdx.x * blockDim.x + threadIdx.x; e < n; e += stride) {
    if (e + stride < n) {
      __builtin_prefetch(gs + e + stride, 0, 0);
      __builtin_prefetch(gd + e + stride, 0, 0);
    }
    size_t si = (size_t)gs[e] * 12u;
    size_t di = (size_t)gd[e] * 12u;
    float4 s0 = *(const float4*)(src + si);
    float4 s1 = *(const float4*)(src + si + 4);
    float4 s2 = *(const float4*)(src + si + 8);
    float4 d0 = *(const float4*)(dst + di);
    float4 d1 = *(const float4*)(dst + di + 4);
    float4 d2 = *(const float4*)(dst + di + 8);
    float acc = s0.x * d0.x;
    acc = fmaf(s0.y, d0.y, acc);
    acc = fmaf(s0.z, d0.z, acc);
    acc = fmaf(s0.w, d0.w, acc);
    acc = fmaf(s1.x, d1.x, acc);
    acc = fmaf(s1.y, d1.y, acc);
    acc = fmaf(s1.z, d1.z, acc);
    acc = fmaf(s1.w, d1.w, acc);
    acc = fmaf(s2.x, d2.x, acc);
    acc = fmaf(s2.y, d2.y, acc);
    acc = fmaf(s2.z, d2.z, acc);
    acc = fmaf(s2.w, d2.w, acc);
    lik[e] = acc;
  }
}

// ---------------------------------------------------------------------------
// K2: batch-norm statistics via V_WMMA_F32_16X16X4_F32.
// B = all-ones => C[m][n] += row-sum of A. One WMMA folds 64 f32 values into
// the f32 accumulator; sum and sum-of-squares chained per 64-elt chunk.
// Main loop is guard-free (single global_load_b64 per lane per chunk); the
// single possible partial chunk is handled by wave 0 with per-lane guards.
// Columns of C are replicated 16x, so wave_sum(per-lane VGPR sums) == 16*total.
// ---------------------------------------------------------------------------
__global__ void dgc_stats_wmma(const float* __restrict__ lik,
                               float* __restrict__ stats, int n) {
  int lane   = threadIdx.x & 31;
  int wave   = (blockIdx.x * blockDim.x + threadIdx.x) >> 5;
  int nwaves = (gridDim.x * blockDim.x) >> 5;
  int nfull  = n >> 6;                 // guard-free 64-element chunks

  v8f csum = {};
  v8f csq  = {};
  v2f ones; ones[0] = 1.0f; ones[1] = 1.0f;

  for (int ch = wave; ch < nfull; ch += nwaves) {
    v2f a = *(const v2f*)(lik + ((size_t)ch << 6) + (lane << 1));
    v2f a2; a2[0] = a[0] * a[0]; a2[1] = a[1] * a[1];
    // (neg_a, A, neg_b, B, c_mod, C, reuse_a, reuse_b)
    csum = __builtin_amdgcn_wmma_f32_16x16x4_f32(false, a,  false, ones,
                                                 (short)0, csum, false, false);
    csq  = __builtin_amdgcn_wmma_f32_16x16x4_f32(false, a2, false, ones,
                                                 (short)0, csq,  false, false);
  }

  // Tail (n % 64 elements): wave-uniform branch; loads guarded per lane,
  // WMMA itself still executes with EXEC all-ones.
  if (wave == 0 && (n & 63)) {
    int base = (nfull << 6) + (lane << 1);
    float x0 = (base < n)     ? lik[base]     : 0.0f;
    float x1 = (base + 1 < n) ? lik[base + 1] : 0.0f;
    v2f a;  a[0]  = x0;      a[1]  = x1;
    v2f a2; a2[0] = x0 * x0; a2[1] = x1 * x1;
    csum = __builtin_amdgcn_wmma_f32_16x16x4_f32(false, a,  false, ones,
                                                 (short)0, csum, false, false);
    csq  = __builtin_amdgcn_wmma_f32_16x16x4_f32(false, a2, false, ones,
                                                 (short)0, csq,  false, false);
  }

  float s = csum[0] + csum[1] + csum[2] + csum[3] +
            csum[4] + csum[5] + csum[6] + csum[7];
  float q = csq[0] + csq[1] + csq[2] + csq[3] +
            csq[4] + csq[5] + csq[6] + csq[7];
  for (int off = 16; off > 0; off >>= 1) {
    s += __shfl_xor(s, off, 32);
    q += __shfl_xor(q, off, 32);
  }
  if (lane == 0) {
    atomicAdd(stats,     s * (1.0f / 16.0f));
    atomicAdd(stats + 1, q * (1.0f / 16.0f));
  }
}

// ---------------------------------------------------------------------------
// K3: fold BN into y = exp(a*x + c):  a = w*rsqrt(var+eps), c = b - mean*a
// ---------------------------------------------------------------------------
__global__ void dgc_finalize(const float* __restrict__ stats,
                             const float* __restrict__ bn_w,
                             const float* __restrict__ bn_b,
                             float* __restrict__ coef, float inv_n) {
  float mean = stats[0] * inv_n;
  float var  = fmaf(-mean, mean, stats[1] * inv_n);
  var = fmaxf(var, 0.0f);
  float a = rsqrtf(var + BN_EPS) * bn_w[0];
  coef[0] = a;
  coef[1] = fmaf(-mean, a, bn_b[0]);
}

// ---------------------------------------------------------------------------
// K4: edge_weight = exp(a*lik + c)  (in place in d_out) + scatter-add denom
// ---------------------------------------------------------------------------
__global__ void dgc_exp_scatter(float* __restrict__ ew,
                                const int* __restrict__ gs,
                                const float* __restrict__ coef,
                                float* __restrict__ denom, int n) {
  float a = coef[0];
  float c = coef[1];
  int stride = gridDim.x * blockDim.x;
  for (int e = blockIdx.x * blockDim.x + threadIdx.x; e < n; e += stride) {
    float w = __expf(fmaf(ew[e], a, c));
    ew[e] = w;
    atomicAdd(denom + gs[e], w);
  }
}

// ---------------------------------------------------------------------------
// K5: per-edge normalization by gathered denominator
// ---------------------------------------------------------------------------
__global__ void dgc_normalize(float* __restrict__ ew,
                              const int* __restrict__ gs,
                              const float* __restrict__ denom, int n) {
  int stride = gridDim.x * blockDim.x;
  for (int e = blockIdx.x * blockDim.x + threadIdx.x; e < n; e += stride) {
    ew[e] = ew[e] / (1e-12f + denom[gs[e]]);
  }
}

extern "C" void kernel_launch(void* const* d_in, const int* in_sizes, int n_in,
                              void* d_out, int out_size, void* d_ws, size_t ws_size,
                              hipStream_t stream) {
  const float* src  = (const float*)d_in[0];
  const float* dst  = (const float*)d_in[1];
  const float* bn_w = (const float*)d_in[2];
  const float* bn_b = (const float*)d_in[3];
  const int*   gs   = (const int*)d_in[4];
  const int*   gd   = (const int*)d_in[5];

  int n_nodes = in_sizes[0] / 12;
  int n_edges = in_sizes[4];
  float* out = (float*)d_out;   // reused: likelihood -> edge weight (in place)

  // workspace: [stats:2f][coef:2f][pad to 256B][denom:n_nodes f]
  char*  ws    = (char*)d_ws;
  float* stats = (float*)ws;
  float* coef  = stats + 2;
  float* denom = (float*)(ws + 256);

  hipMemsetAsync(stats, 0, 2 * sizeof(float), stream);
  hipMemsetAsync(denom, 0, (size_t)n_nodes * sizeof(float), stream);

  const int block = 256;                    // 8 wave32 per block
  int grid  = (n_edges + block - 1) / block;
  if (grid > 2048) grid = 2048;             // grid-stride cap

  dgc_edge_dot<<<grid, block, 0, stream>>>(src, dst, gs, gd, out, n_edges);
  dgc_stats_wmma<<<128, block, 0, stream>>>(out, stats, n_edges);
  dgc_finalize<<<1, 1, 0, stream>>>(stats, bn_w, bn_b, coef,
                                    1.0f / (float)n_edges);
  dgc_exp_scatter<<<grid, block, 0, stream>>>(out, gs, coef, denom, n_edges);
  dgc_normalize<<<grid, block, 0, stream>>>(out, gs, denom, n_edges);
}